// GCN_10642928960106
// MI455X (gfx1250) — compile-verified
//
#include <hip/hip_runtime.h>
#include <hip/hip_bf16.h>

typedef __attribute__((ext_vector_type(2))) float v2f;
typedef __attribute__((ext_vector_type(8))) float v8f;

#define F 128
#define NG 64
#define NC 10

// ---------------- degree / normalization ----------------

__global__ void k_init_deg(float* __restrict__ deg, int N) {
    int v = blockIdx.x * blockDim.x + threadIdx.x;
    if (v < N) deg[v] = 1.0f;                       // self-loop weight
}

__global__ void k_deg_edges(const int* __restrict__ col, float* __restrict__ deg, int E) {
    int e = blockIdx.x * blockDim.x + threadIdx.x;
    if (e < E) atomicAdd(&deg[col[e]], 1.0f);
}

__global__ void k_dis(const float* __restrict__ deg, float* __restrict__ dis, int N) {
    int v = blockIdx.x * blockDim.x + threadIdx.x;
    if (v < N) dis[v] = rsqrtf(deg[v]);             // deg >= 1 always
}

// ---------------- WMMA fp32 GEMM: Hw = A(MxF) @ W(FxF), Acc = Hw * dis^2 ----------------
// Block = 256 threads = 8 waves. Wave w computes the 16x16 tile at (blockIdx.x*16, w*16).
// K=128 consumed as 32 steps of V_WMMA_F32_16X16X4_F32.

__global__ __launch_bounds__(256) void k_gemm_selfinit(
    const float* __restrict__ A, const float* __restrict__ W,
    const float* __restrict__ dis, float* __restrict__ Hw,
    float* __restrict__ Acc, int M) {
    const int wave  = threadIdx.x >> 5;
    const int lane  = threadIdx.x & 31;
    const int mBase = blockIdx.x << 4;
    const int nBase = wave << 4;
    const int laneHi = lane >> 4;          // 0 or 1
    const int kOff   = laneHi << 1;        // K sub-offset 0 or 2 (ISA 16x4 f32 A layout)
    const int mRow   = lane & 15;
    const int n      = lane & 15;

    // clamped row so EXEC stays all-ones even at a ragged tail (M=40000 is exact anyway)
    int mA = mBase + mRow; if (mA > M - 1) mA = M - 1;
    const float* __restrict__ arow = A + (size_t)mA * F;
    const float* __restrict__ wcol = W + nBase + n;

    v8f c = {0.f, 0.f, 0.f, 0.f, 0.f, 0.f, 0.f, 0.f};
    #pragma unroll 8
    for (int k = 0; k < F; k += 4) {
        v2f a, b;
        a[0] = arow[k + kOff];
        a[1] = arow[k + kOff + 1];
        b[0] = wcol[(size_t)(k + kOff) * F];
        b[1] = wcol[(size_t)(k + kOff + 1) * F];
        // D = A*B + C  (fp32, RNE — matches fp32 reference up to summation order)
        c = __builtin_amdgcn_wmma_f32_16x16x4_f32(false, a, false, b,
                                                  (short)0, c, false, false);
    }

    // Epilogue. mBase is wave-uniform -> uniform branch; full tiles (the only
    // case when M%16==0) take the branch-free, fully coalesced store path.
    const int mHi = laneHi << 3;           // C/D layout: VGPR r -> M = r + 8*laneHi
    if (mBase + 16 <= M) {
        #pragma unroll
        for (int r = 0; r < 8; ++r) {
            const int m = mBase + mHi + r;
            const float d = dis[m];
            const size_t idx = (size_t)m * F + nBase + n;
            Hw[idx]  = c[r];
            Acc[idx] = c[r] * d * d;       // self-loop contribution: norm = dis[v]^2
        }
    } else {
        #pragma unroll
        for (int r = 0; r < 8; ++r) {
            const int m = mBase + mHi + r;
            if (m < M) {
                const float d = dis[m];
                const size_t idx = (size_t)m * F + nBase + n;
                Hw[idx]  = c[r];
                Acc[idx] = c[r] * d * d;
            }
        }
    }
}

// ---------------- edge scatter: Acc[col] += Hw[row] * dis[row]*dis[col] ----------------
// 1 thread per (edge, 4-float chunk): float4 gather + 4 atomic f32 adds (L2-resident).

__global__ __launch_bounds__(256) void k_scatter(
    const int* __restrict__ row, const int* __restrict__ col,
    const float* __restrict__ dis, const float* __restrict__ Hw,
    float* __restrict__ Acc, int E) {
    long long t = (long long)blockIdx.x * 256 + threadIdx.x;
    int e = (int)(t >> 5);
    if (e >= E) return;
    int ck = ((int)t & 31) << 2;
    int r = row[e], c = col[e];
    float nrm = dis[r] * dis[c];
    const float4 h = *(const float4*)(Hw + (size_t)r * F + ck);
    float* o = Acc + (size_t)c * F + ck;
    atomicAdd(o + 0, h.x * nrm);
    atomicAdd(o + 1, h.y * nrm);
    atomicAdd(o + 2, h.z * nrm);
    atomicAdd(o + 3, h.w * nrm);
}

__global__ void k_bias_relu(float* __restrict__ Acc, const float* __restrict__ b, long long n) {
    long long t = (long long)blockIdx.x * blockDim.x + threadIdx.x;
    if (t < n) Acc[t] = fmaxf(Acc[t] + b[(int)t & (F - 1)], 0.0f);
}

// ---------------- global mean pool + final linear ----------------

__global__ void k_pool_zero(float* __restrict__ pooled, float* __restrict__ cnt) {
    int t = blockIdx.x * blockDim.x + threadIdx.x;
    if (t < NG * F) pooled[t] = 0.0f;
    if (t < NG)     cnt[t] = 0.0f;
}

__global__ __launch_bounds__(256) void k_pool_acc(
    const int* __restrict__ batch, const float* __restrict__ H,
    float* __restrict__ pooled, float* __restrict__ cnt, int N) {
    long long t = (long long)blockIdx.x * 256 + threadIdx.x;
    int v = (int)(t >> 5);
    if (v >= N) return;
    int ck = ((int)t & 31) << 2;
    int g = batch[v];
    const float4 h = *(const float4*)(H + (size_t)v * F + ck);
    float* p = pooled + (size_t)g * F + ck;
    atomicAdd(p + 0, h.x);
    atomicAdd(p + 1, h.y);
    atomicAdd(p + 2, h.z);
    atomicAdd(p + 3, h.w);
    if (((int)t & 31) == 0) atomicAdd(&cnt[g], 1.0f);
}

__global__ void k_final(const float* __restrict__ pooled, const float* __restrict__ cnt,
                        const float* __restrict__ Wl, const float* __restrict__ bl,
                        float* __restrict__ out) {
    int t = blockIdx.x * blockDim.x + threadIdx.x;
    if (t >= NG * NC) return;
    int g = t / NC, c = t % NC;
    float inv = 1.0f / fmaxf(cnt[g], 1.0f);
    float acc = bl[c];
    #pragma unroll 8
    for (int f = 0; f < F; ++f)
        acc += pooled[g * F + f] * inv * Wl[f * NC + c];
    out[t] = acc;
}

// ---------------- launcher ----------------

extern "C" void kernel_launch(void* const* d_in, const int* in_sizes, int n_in,
                              void* d_out, int out_size, void* d_ws, size_t ws_size,
                              hipStream_t stream) {
    const float* x     = (const float*)d_in[0];
    const int*   ei    = (const int*)  d_in[1];
    const int*   batch = (const int*)  d_in[2];
    const float* W1 = (const float*)d_in[3];  const float* b1 = (const float*)d_in[4];
    const float* W2 = (const float*)d_in[5];  const float* b2 = (const float*)d_in[6];
    const float* W3 = (const float*)d_in[7];  const float* b3 = (const float*)d_in[8];
    const float* Wl = (const float*)d_in[9];  const float* bl = (const float*)d_in[10];

    const int N = in_sizes[0] / F;
    const int E = in_sizes[1] / 2;
    const int* row = ei;
    const int* col = ei + E;

    // workspace carve: deg[N] dis[N] Hw[N*F] bufA[N*F] bufB[N*F] pooled[NG*F] cnt[NG]
    float* ws     = (float*)d_ws;
    float* deg    = ws;
    float* dis    = deg + N;
    float* Hw     = dis + N;
    float* bufA   = Hw   + (size_t)N * F;
    float* bufB   = bufA + (size_t)N * F;
    float* pooled = bufB + (size_t)N * F;
    float* cnt    = pooled + NG * F;
    float* outp   = (float*)d_out;

    const int TB = 256;
    const unsigned gN  = (unsigned)((N + TB - 1) / TB);
    const unsigned gE  = (unsigned)((E + TB - 1) / TB);
    const unsigned gM  = (unsigned)((N + 15) / 16);              // GEMM row-tiles
    const long long ne = (long long)N * F;
    const unsigned gNE = (unsigned)((ne + TB - 1) / TB);
    const unsigned gSc = (unsigned)(((long long)E * 32 + TB - 1) / TB);
    const unsigned gPl = (unsigned)(((long long)N * 32 + TB - 1) / TB);

    // normalization
    k_init_deg <<<gN, TB, 0, stream>>>(deg, N);
    k_deg_edges<<<gE, TB, 0, stream>>>(col, deg, E);
    k_dis      <<<gN, TB, 0, stream>>>(deg, dis, N);

    // layer 1: x -> bufA
    k_gemm_selfinit<<<gM, 256, 0, stream>>>(x, W1, dis, Hw, bufA, N);
    k_scatter      <<<gSc, 256, 0, stream>>>(row, col, dis, Hw, bufA, E);
    k_bias_relu    <<<gNE, TB, 0, stream>>>(bufA, b1, ne);

    // layer 2: bufA -> bufB
    k_gemm_selfinit<<<gM, 256, 0, stream>>>(bufA, W2, dis, Hw, bufB, N);
    k_scatter      <<<gSc, 256, 0, stream>>>(row, col, dis, Hw, bufB, E);
    k_bias_relu    <<<gNE, TB, 0, stream>>>(bufB, b2, ne);

    // layer 3: bufB -> bufA
    k_gemm_selfinit<<<gM, 256, 0, stream>>>(bufB, W3, dis, Hw, bufA, N);
    k_scatter      <<<gSc, 256, 0, stream>>>(row, col, dis, Hw, bufA, E);
    k_bias_relu    <<<gNE, TB, 0, stream>>>(bufA, b3, ne);

    // pool + classify
    k_pool_zero<<<(NG * F + TB - 1) / TB, TB, 0, stream>>>(pooled, cnt);
    k_pool_acc <<<gPl, 256, 0, stream>>>(batch, bufA, pooled, cnt, N);
    k_final    <<<(NG * NC + TB - 1) / TB, TB, 0, stream>>>(pooled, cnt, Wl, bl, outp);
}